// EmgLoss_8237747274455
// MI455X (gfx1250) — compile-verified
//
#include <hip/hip_runtime.h>
#include <math.h>

// Problem constants (match reference: B=8, T=S=1024, C=16, eps=0.05^2, 50 iters)
#define BB 8
#define TT 1024
#define CC 16
#define EPS_F   0.0025f
#define INV_EPS 400.0f
#define NITERS  50
#define LOGW    (-6.93147180559945309f)   // -log(1024)
#define POT_STRIDE (BB*TT)                // 8192 floats per potential
#define BPAD 18                           // padded LDS row stride (floats), conflict-free & 8B aligned

typedef __attribute__((ext_vector_type(2))) float v2f;
typedef __attribute__((ext_vector_type(8))) float v8f;

// ---------------------------------------------------------------------------
// Setup: squared norms of every point + MSE partial sums.
// One thread per (b, t) row; 8192 rows total.
// ---------------------------------------------------------------------------
__global__ void __launch_bounds__(256)
sink_setup(const float* __restrict__ x, const float* __restrict__ y,
           float* __restrict__ x2, float* __restrict__ y2,
           float* __restrict__ mse_acc)
{
    int t = blockIdx.x * blockDim.x + threadIdx.x;   // 0..8191
    float msep = 0.f;
    if (t < BB * TT) {
        const float* xr = x + (size_t)t * CC;
        const float* yr = y + (size_t)t * CC;
        float sx = 0.f, sy = 0.f;
        #pragma unroll
        for (int c = 0; c < CC; ++c) {
            float xv = xr[c], yv = yr[c];
            sx += xv * xv;
            sy += yv * yv;
            float d = xv - yv;
            msep += d * d;
        }
        x2[t] = sx;
        y2[t] = sy;
    }
    // block reduction of the MSE partial
    __shared__ float red[8];
    #pragma unroll
    for (int off = 16; off > 0; off >>= 1) msep += __shfl_down(msep, off, 32);
    int lane = threadIdx.x & 31, wv = threadIdx.x >> 5;
    if (lane == 0) red[wv] = msep;
    __syncthreads();
    if (threadIdx.x == 0) {
        float s = 0.f;
        #pragma unroll
        for (int i = 0; i < 8; ++i) s += red[i];
        atomicAdd(mse_acc, s);      // global_atomic_add_f32
    }
}

// ---------------------------------------------------------------------------
// One Sinkhorn half-step: computes all four softmin updates (f,g,p,q),
// averaged with the old potentials (geomloss symmetric stabilized update).
//
// For each update:  out_i = 0.5*(old_i + (-eps*(rowterm_i + LSE_j(colterm_j + A_i.B_j/eps))))
// The dot products A_i.B_j are produced by V_WMMA_F32_16X16X4_F32 (K=16 via
// 4 chained WMMAs); A rows are pre-scaled by 1/eps at register-load time.
//
// grid.x = 4 updates * 8 batches * 8 row-groups = 256 blocks
// block  = 256 threads = 8 waves; wave w owns rows [ (grp*8+w)*16 , +16 )
// LDS: B matrix of this batch, padded stride 18, + 1024 column terms.
// ---------------------------------------------------------------------------
__global__ void __launch_bounds__(256)
sinkhorn_step(const float* __restrict__ x, const float* __restrict__ y,
              const float* __restrict__ x2, const float* __restrict__ y2,
              const float* __restrict__ pin, float* __restrict__ pout)
{
    extern __shared__ float smem[];          // TT*BPAD + TT floats
    float* Blds = smem;                      // [TT][BPAD]
    float* CT   = smem + TT * BPAD;          // [TT] column terms

    const int u   = blockIdx.x >> 6;         // update selector 0..3
    const int b   = (blockIdx.x >> 3) & 7;   // batch
    const int grp = blockIdx.x & 7;          // row-group

    const float* f = pin + 0 * POT_STRIDE;
    const float* g = pin + 1 * POT_STRIDE;
    const float* p = pin + 2 * POT_STRIDE;
    const float* q = pin + 3 * POT_STRIDE;

    const float *Amat, *Bmat, *A2, *W2, *h, *oldp;
    float* outp;
    switch (u) {
        case 0:  Amat = x; Bmat = y; A2 = x2; W2 = y2; h = g; oldp = f; outp = pout + 0 * POT_STRIDE; break; // f1
        case 1:  Amat = y; Bmat = x; A2 = y2; W2 = x2; h = f; oldp = g; outp = pout + 1 * POT_STRIDE; break; // g1
        case 2:  Amat = x; Bmat = x; A2 = x2; W2 = x2; h = p; oldp = p; outp = pout + 2 * POT_STRIDE; break; // p1
        default: Amat = y; Bmat = y; A2 = y2; W2 = y2; h = q; oldp = q; outp = pout + 3 * POT_STRIDE; break; // q1
    }

    // --- cooperative load of B (this batch) into padded LDS ---
    const float* Bb = Bmat + (size_t)b * TT * CC;
    for (int r = threadIdx.x; r < TT; r += 256) {
        const float4* src = (const float4*)(Bb + (size_t)r * CC);
        float4 v0 = src[0], v1 = src[1], v2 = src[2], v3 = src[3];
        float2* dst = (float2*)(Blds + r * BPAD);      // r*BPAD even -> 8B aligned
        dst[0] = make_float2(v0.x, v0.y); dst[1] = make_float2(v0.z, v0.w);
        dst[2] = make_float2(v1.x, v1.y); dst[3] = make_float2(v1.z, v1.w);
        dst[4] = make_float2(v2.x, v2.y); dst[5] = make_float2(v2.z, v2.w);
        dst[6] = make_float2(v3.x, v3.y); dst[7] = make_float2(v3.z, v3.w);
    }
    // column terms: logw + (h_j - 0.5*||B_j||^2)/eps
    for (int j = threadIdx.x; j < TT; j += 256)
        CT[j] = LOGW + (h[b * TT + j] - 0.5f * W2[b * TT + j]) * INV_EPS;
    __syncthreads();

    const int lane = threadIdx.x & 31;
    const int wv   = threadIdx.x >> 5;
    const int l15  = lane & 15;
    const int hi   = lane >> 4;                    // 0: K lanes 0/1, rows 0-7 of D ; 1: K lanes 2/3, rows 8-15
    const int rbase = (grp * 8 + wv) * 16;

    // A fragments (16x4 f32 x 4 chunks), pre-scaled by 1/eps.
    // Layout: lanes 0-15 hold M=lane K={k,k+1}; lanes 16-31 hold M=lane-16 K={k+2,k+3}.
    const float* Arow = Amat + ((size_t)b * TT + rbase + l15) * CC;
    v2f afrag[4];
    #pragma unroll
    for (int kc = 0; kc < 4; ++kc) {
        int k0 = kc * 4 + hi * 2;
        afrag[kc].x = Arow[k0 + 0] * INV_EPS;
        afrag[kc].y = Arow[k0 + 1] * INV_EPS;
    }

    // Online logsumexp state: 8 rows per lane (rows rbase + hi*8 + r, partial over
    // the columns this lane sees: j = 16*tile + l15).
    float m[8], s[8];
    #pragma unroll
    for (int r = 0; r < 8; ++r) { m[r] = -INFINITY; s[r] = 0.f; }

    for (int tile = 0; tile < TT / 16; ++tile) {
        const int jbase = tile * 16;
        // B fragment: element (k, n) = Bmat[jbase+n][k]; mirrors A layout.
        const float* brow = Blds + (jbase + l15) * BPAD + hi * 2;
        v8f c = {0.f, 0.f, 0.f, 0.f, 0.f, 0.f, 0.f, 0.f};
        #pragma unroll
        for (int kc = 0; kc < 4; ++kc) {
            v2f bfrag;
            bfrag.x = brow[kc * 4 + 0];
            bfrag.y = brow[kc * 4 + 1];
            c = __builtin_amdgcn_wmma_f32_16x16x4_f32(
                    false, afrag[kc], false, bfrag, (short)0, c, false, false);
        }
        const float ct = CT[jbase + l15];
        #pragma unroll
        for (int r = 0; r < 8; ++r) {
            float v  = c[r] + ct;
            float nm = fmaxf(m[r], v);
            s[r] = s[r] * __expf(m[r] - nm) + __expf(v - nm);
            m[r] = nm;
        }
    }

    // Merge (m,s) across the 16 lanes that share each row (stay inside 16-lane half).
    #pragma unroll
    for (int r = 0; r < 8; ++r) {
        #pragma unroll
        for (int off = 1; off < 16; off <<= 1) {
            float om = __shfl_xor(m[r], off, 32);
            float os = __shfl_xor(s[r], off, 32);
            float nm = fmaxf(m[r], om);
            s[r] = s[r] * __expf(m[r] - nm) + os * __expf(om - nm);
            m[r] = nm;
        }
    }

    if (l15 == 0) {
        #pragma unroll
        for (int r = 0; r < 8; ++r) {
            int row = rbase + hi * 8 + r;
            float lse = m[r] + __logf(s[r]);
            float rowterm = -0.5f * A2[b * TT + row] * INV_EPS;
            float res = -EPS_F * (rowterm + lse);
            outp[b * TT + row] = 0.5f * (oldp[b * TT + row] + res);
        }
    }
}

// ---------------------------------------------------------------------------
// Finalize: ot = mean_b[ mean_i(f-p) + mean_j(g-q) ], combine with MSE.
// ---------------------------------------------------------------------------
__global__ void __launch_bounds__(256)
sink_finalize(const float* __restrict__ pots, const float* __restrict__ mse_acc,
              float* __restrict__ out)
{
    const float* f = pots + 0 * POT_STRIDE;
    const float* g = pots + 1 * POT_STRIDE;
    const float* p = pots + 2 * POT_STRIDE;
    const float* q = pots + 3 * POT_STRIDE;
    float sum = 0.f;
    for (int i = threadIdx.x; i < POT_STRIDE; i += 256)
        sum += (f[i] - p[i]) + (g[i] - q[i]);
    __shared__ float red[8];
    #pragma unroll
    for (int off = 16; off > 0; off >>= 1) sum += __shfl_down(sum, off, 32);
    int lane = threadIdx.x & 31, wv = threadIdx.x >> 5;
    if (lane == 0) red[wv] = sum;
    __syncthreads();
    if (threadIdx.x == 0) {
        float tot = 0.f;
        #pragma unroll
        for (int i = 0; i < 8; ++i) tot += red[i];
        float ot  = tot / (float)POT_STRIDE;                 // /(B*T) since loga=logb uniform
        float mse = mse_acc[0] / (float)(BB * TT * CC);
        out[0] = mse + ot;   // total (W_MSE=W_OT=1)
        out[1] = mse;
        out[2] = ot;
    }
}

// ---------------------------------------------------------------------------
// Host launcher.
// Workspace (floats):
//   [0,8192)            x2
//   [8192,16384)        y2
//   [16384,16384+65536) potentials, 2 parities x 4 pots x 8192
//   [81920]             mse accumulator
// ---------------------------------------------------------------------------
extern "C" void kernel_launch(void* const* d_in, const int* in_sizes, int n_in,
                              void* d_out, int out_size, void* d_ws, size_t ws_size,
                              hipStream_t stream)
{
    const float* x = (const float*)d_in[0];   // pred_sequence (8,1024,16) f32
    const float* y = (const float*)d_in[1];   // gt_sequence
    float* ws   = (float*)d_ws;
    float* x2   = ws;
    float* y2   = ws + 8192;
    float* pots = ws + 16384;                 // 2 * 4 * 8192 floats
    float* msea = ws + 16384 + 2 * 4 * POT_STRIDE;

    // zero initial potentials (parity 0) and the MSE accumulator (capturable)
    hipMemsetAsync(pots, 0, (size_t)4 * POT_STRIDE * sizeof(float), stream);
    hipMemsetAsync(msea, 0, sizeof(float), stream);

    sink_setup<<<32, 256, 0, stream>>>(x, y, x2, y2, msea);

    const size_t shmem = (size_t)(TT * BPAD + TT) * sizeof(float);  // ~76KB, fits 320KB WGP LDS
    for (int it = 0; it < NITERS; ++it) {
        float* pin  = pots + (size_t)(it & 1) * 4 * POT_STRIDE;
        float* pout = pots + (size_t)((it + 1) & 1) * 4 * POT_STRIDE;
        sinkhorn_step<<<256, 256, shmem, stream>>>(x, y, x2, y2, pin, pout);
    }
    // NITERS even -> final potentials in parity 0
    sink_finalize<<<1, 256, 0, stream>>>(pots + (size_t)(NITERS & 1) * 4 * POT_STRIDE,
                                         msea, (float*)d_out);
}